// GNNLayer_54099408060613
// MI455X (gfx1250) — compile-verified
//
#include <hip/hip_runtime.h>

typedef __attribute__((ext_vector_type(2))) float v2f;
typedef __attribute__((ext_vector_type(8))) float v8f;
typedef __attribute__((__vector_size__(16))) int b128_t;
typedef __attribute__((address_space(1))) b128_t* gptr_b128;
typedef __attribute__((address_space(3))) b128_t* lptr_b128;

#define IN_F 512
#define OUT_F 128
#define ROWS_PB 32          // feature rows per block (100000 % 32 == 0 -> 3125 blocks)
#define KC 128              // K-chunk staged in LDS (4 chunks, double buffered)
#define LDS_STRIDE (KC + 4) // 132 floats: row = 528B (16B aligned), 4-bank skew per row

// ---- CDNA5 async global->LDS copy (ASYNCcnt path, no VGPR round trip) ------
__device__ __forceinline__ void async_copy_b128(const float* g, float* l) {
#if __has_builtin(__builtin_amdgcn_global_load_async_to_lds_b128)
    __builtin_amdgcn_global_load_async_to_lds_b128(
        (gptr_b128)(void*)g,   // global source (addrspace 1)
        (lptr_b128)(void*)l,   // LDS destination (addrspace 3)
        0, 0);
#else
    // vdst = LDS byte address (low 32 bits of generic LDS pointer), vaddr = 64b
    asm volatile("global_load_async_to_lds_b128 %0, %1, off"
                 :
                 : "v"((unsigned int)(unsigned long long)l), "v"(g)
                 : "memory");
#endif
}

__device__ __forceinline__ void wait_async0() {
#if __has_builtin(__builtin_amdgcn_s_wait_asynccnt)
    __builtin_amdgcn_s_wait_asynccnt(0);
#else
    asm volatile("s_wait_asynccnt 0x0" ::: "memory");
#endif
}

__device__ __forceinline__ void stage_chunk_async(const float* fbase, int kc,
                                                  float* dst, int tid) {
    // 32 rows x 128 cols fp32 = 1024 x b128; 256 threads -> 4 async copies each
#pragma unroll
    for (int i = tid; i < (ROWS_PB * KC) / 4; i += 256) {
        const int e = i * 4;
        const int r = e / KC;
        const int c = e % KC;
        async_copy_b128(fbase + (size_t)r * IN_F + kc + c, dst + r * LDS_STRIDE + c);
    }
}

// ---------------------------------------------------------------------------
// support = features @ W   (fp32 WMMA 16x16x4, double-buffered async staging)
// block: 256 threads = 8 waves; wave w -> N-tile w (cols 16w..16w+15)
// block handles 32 rows -> each wave accumulates two 16x16 tiles
// ---------------------------------------------------------------------------
__global__ __launch_bounds__(256) void gnn_gemm_kernel(const float* __restrict__ feat,
                                                       const float* __restrict__ W,
                                                       float* __restrict__ sup) {
    __shared__ float lds[2][ROWS_PB * LDS_STRIDE];

    const int tid  = threadIdx.x;
    const int m0   = blockIdx.x * ROWS_PB;
    const int wave = tid >> 5;       // 0..7 = N tile
    const int lane = tid & 31;
    const int half = lane >> 4;      // 0/1 (lane group)
    const int ln16 = lane & 15;
    const int ncol = wave * 16 + ln16;
    const float* fbase = feat + (size_t)m0 * IN_F;

    v8f acc0 = {};
    v8f acc1 = {};

    // prologue: chunk 0 -> buffer 0
    stage_chunk_async(fbase, 0, lds[0], tid);
    wait_async0();
    __syncthreads();

    for (int kc = 0; kc < IN_F; kc += KC) {
        const int buf      = (kc / KC) & 1;
        const int has_next = (kc + KC < IN_F);

        // kick off next chunk into the other buffer; overlaps with WMMA below
        if (has_next)
            stage_chunk_async(fbase, kc + KC, lds[buf ^ 1], tid);

        const float* lbuf = lds[buf];
#pragma unroll 8
        for (int k = 0; k < KC; k += 4) {
            // B fragment: 4x16 fp32. Lane half h holds K = {2h, 2h+1} in vgpr 0/1.
            v2f b;
            b.x = W[(size_t)(kc + k + 2 * half + 0) * OUT_F + ncol];
            b.y = W[(size_t)(kc + k + 2 * half + 1) * OUT_F + ncol];
            // A fragments: 16x4 fp32, lane (0..15)=M, half h holds K = {2h, 2h+1}
            const v2f a0 = *(const v2f*)(&lbuf[(0 * 16 + ln16) * LDS_STRIDE + k + 2 * half]);
            const v2f a1 = *(const v2f*)(&lbuf[(1 * 16 + ln16) * LDS_STRIDE + k + 2 * half]);
            acc0 = __builtin_amdgcn_wmma_f32_16x16x4_f32(false, a0, false, b,
                                                         (short)0, acc0, false, false);
            acc1 = __builtin_amdgcn_wmma_f32_16x16x4_f32(false, a1, false, b,
                                                         (short)0, acc1, false, false);
        }

        if (has_next) {
            wait_async0();     // my async copies have landed in LDS
            __syncthreads();   // everyone done reading lds[buf] + all copies visible
        }
    }

    // C/D layout: vgpr g, lanes 0-15 -> M=g, lanes 16-31 -> M=g+8, N = lane%16
#pragma unroll
    for (int g = 0; g < 8; ++g) {
        const int r0 = m0 + 0 * 16 + g + 8 * half;
        const int r1 = m0 + 1 * 16 + g + 8 * half;
        sup[(size_t)r0 * OUT_F + ncol] = acc0[g];
        sup[(size_t)r1 * OUT_F + ncol] = acc1[g];
    }
}

// ---------------------------------------------------------------------------
// Edge-parallel SpMM: one wave32 per edge; lane owns 4 contiguous cols.
// support/out are L2-resident (51 MB each << 192 MB L2) -> gathers + fp32
// atomics run at L2, not HBM.
// ---------------------------------------------------------------------------
__global__ __launch_bounds__(256) void gnn_spmm_kernel(const float* __restrict__ sup,
                                                       const int* __restrict__ esrc,
                                                       const int* __restrict__ edst,
                                                       const float* __restrict__ ew,
                                                       float* __restrict__ out,
                                                       int n_edges) {
    const int gwave = (int)((blockIdx.x * 256u + threadIdx.x) >> 5);
    const int lane  = threadIdx.x & 31;
    if (gwave >= n_edges) return;

    const int   s = esrc[gwave];
    const int   d = edst[gwave];
    const float w = ew[gwave];

    const float4 v = ((const float4*)(sup + (size_t)s * OUT_F))[lane];
    float* o = out + (size_t)d * OUT_F + lane * 4;

    __hip_atomic_fetch_add(o + 0, v.x * w, __ATOMIC_RELAXED, __HIP_MEMORY_SCOPE_AGENT);
    __hip_atomic_fetch_add(o + 1, v.y * w, __ATOMIC_RELAXED, __HIP_MEMORY_SCOPE_AGENT);
    __hip_atomic_fetch_add(o + 2, v.z * w, __ATOMIC_RELAXED, __HIP_MEMORY_SCOPE_AGENT);
    __hip_atomic_fetch_add(o + 3, v.w * w, __ATOMIC_RELAXED, __HIP_MEMORY_SCOPE_AGENT);
}

__global__ __launch_bounds__(256) void gnn_zero_kernel(float* __restrict__ p, int n) {
    for (int i = blockIdx.x * 256 + threadIdx.x; i < n; i += gridDim.x * 256)
        p[i] = 0.0f;
}

__global__ __launch_bounds__(256) void gnn_relu_kernel(float* __restrict__ p, int n) {
    for (int i = blockIdx.x * 256 + threadIdx.x; i < n; i += gridDim.x * 256) {
        const float v = p[i];
        p[i] = v > 0.0f ? v : 0.0f;
    }
}

extern "C" void kernel_launch(void* const* d_in, const int* in_sizes, int n_in,
                              void* d_out, int out_size, void* d_ws, size_t ws_size,
                              hipStream_t stream) {
    const float* feat = (const float*)d_in[0];   // [N, 512]
    const float* W    = (const float*)d_in[1];   // [512, 128]
    const int*   esrc = (const int*)d_in[2];     // [E]
    const int*   edst = (const int*)d_in[3];     // [E]
    const float* ew   = (const float*)d_in[4];   // [E]
    float*       out  = (float*)d_out;           // [N, 128]
    float*       sup  = (float*)d_ws;            // scratch: support [N, 128] fp32

    const int n_nodes = in_sizes[0] / IN_F;      // 100000
    const int n_edges = in_sizes[2];             // 1600000
    const int n_out   = n_nodes * OUT_F;         // 12.8M floats

    // 1) zero accumulator (harness poisons d_out)
    gnn_zero_kernel<<<2048, 256, 0, stream>>>(out, n_out);

    // 2) support = features @ W  (fp32 WMMA, async-to-LDS staging)
    gnn_gemm_kernel<<<n_nodes / ROWS_PB, 256, 0, stream>>>(feat, W, sup);

    // 3) scatter-add messages (one wave per edge, 8 waves per block)
    gnn_spmm_kernel<<<(n_edges + 7) / 8, 256, 0, stream>>>(sup, esrc, edst, ew, out, n_edges);

    // 4) in-place ReLU
    gnn_relu_kernel<<<2048, 256, 0, stream>>>(out, n_out);
}